// FGPillarMaxPooling_44985487458601
// MI455X (gfx1250) — compile-verified
//
#include <hip/hip_runtime.h>

typedef __attribute__((ext_vector_type(2))) float v2f;
typedef __attribute__((ext_vector_type(8))) float v8f;

// Problem constants (match reference config)
#define PCR_X0 (-51.2f)
#define PCR_Y0 (-51.2f)
#define CZ     (-1.0f)        // 0.5*(-5.0 + 3.0)
#define PS     (0.2f)
#define GW     512
#define GH     512
#define COUT   32

__global__ void zero_f4_kernel(float4* __restrict__ out, int n4) {
    int i = blockIdx.x * blockDim.x + threadIdx.x;
    int stride = gridDim.x * blockDim.x;
    for (; i < n4; i += stride) out[i] = make_float4(0.f, 0.f, 0.f, 0.f);
}

__global__ void pillar_wmma_kernel(const float* __restrict__ xyz,
                                   const int*   __restrict__ batch_cnt,
                                   const float* __restrict__ pfeat,
                                   const float* __restrict__ W,
                                   float*       __restrict__ out,
                                   int N, int num_groups) {
    const int lane = threadIdx.x & 31;
    const int half = lane >> 4;     // 0: lanes 0-15, 1: lanes 16-31
    const int mcol = lane & 15;     // M (for A/seg) or N (for B/C) index
    const int waves_per_blk = blockDim.x >> 5;
    const int wave   = blockIdx.x * waves_per_blk + (threadIdx.x >> 5);
    const int nwaves = gridDim.x * waves_per_blk;

    // ---- Load B tiles of W (7x32, padded K->8) into WMMA 4x16 layout ----
    // lane holds B[k = 4*ks + 2*half + vgpr][n = t*16 + mcol]
    v2f Bt[2][2];
#pragma unroll
    for (int ks = 0; ks < 2; ++ks) {
#pragma unroll
        for (int t = 0; t < 2; ++t) {
            int n  = t * 16 + mcol;
            int k0 = ks * 4 + half * 2;
            float b0 = (k0     < 7) ? W[k0 * COUT + n]       : 0.f;
            float b1 = (k0 + 1 < 7) ? W[(k0 + 1) * COUT + n] : 0.f;
            v2f b; b[0] = b0; b[1] = b1;
            Bt[ks][t] = b;
        }
    }

    // batch prefix (B == 4 in this problem)
    const int c0 = batch_cnt[0];
    const int s1 = c0 + batch_cnt[1];
    const int s2 = s1 + batch_cnt[2];

    // per-lane output column element offsets (hoisted out of the atomic branches)
    const long col_lo = (long)mcol;
    const long col_hi = (long)(16 + mcol);

    for (int g = wave; g < num_groups; g += nwaves) {
        const int p     = g * 16 + mcol;       // both half-lanes read point mcol
        const bool valid = (p < N);
        const int pp    = valid ? p : 0;

        const float x = xyz[pp * 3 + 0];
        const float y = xyz[pp * 3 + 1];
        const float z = xyz[pp * 3 + 2];
        const float4 f = *(const float4*)(pfeat + (size_t)pp * 4);

        int px = (int)floorf((x - PCR_X0) / PS); px = min(max(px, 0), GW - 1);
        int py = (int)floorf((y - PCR_Y0) / PS); py = min(max(py, 0), GH - 1);
        const float cx = (px + 0.5f) * PS + PCR_X0;
        const float cy = (py + 0.5f) * PS + PCR_Y0;
        const float ox = x - cx, oy = y - cy, oz = z - CZ;

        const int b   = (pp < c0) ? 0 : (pp < s1) ? 1 : (pp < s2) ? 2 : 3;
        const int seg = b * (GH * GW) + py * GW + px;   // unused rows emit h==0

        // ---- A tiles: lane holds A[m=mcol][k = 2*half + vgpr (+4 for step 1)]
        // Invalid points get zero rows -> D rows are exactly 0 -> no atomics fire.
        v2f A0, A1;
        if (half == 0) { A0[0] = f.x; A0[1] = f.y; A1[0] = ox;  A1[1] = oy; }
        else           { A0[0] = f.z; A0[1] = f.w; A1[0] = oz;  A1[1] = 0.f; }
        if (!valid)    { A0[0] = 0.f; A0[1] = 0.f; A1[0] = 0.f; A1[1] = 0.f; }

        // ---- D = A(16x8) @ W(8x32) via 4x V_WMMA_F32_16X16X4_F32 ----
        v8f c_lo = {}; v8f c_hi = {};
        c_lo = __builtin_amdgcn_wmma_f32_16x16x4_f32(false, A0, false, Bt[0][0], (short)0, c_lo, false, false);
        c_lo = __builtin_amdgcn_wmma_f32_16x16x4_f32(false, A1, false, Bt[1][0], (short)0, c_lo, false, false);
        c_hi = __builtin_amdgcn_wmma_f32_16x16x4_f32(false, A0, false, Bt[0][1], (short)0, c_hi, false, false);
        c_hi = __builtin_amdgcn_wmma_f32_16x16x4_f32(false, A1, false, Bt[1][1], (short)0, c_hi, false, false);

        // ---- scatter-max: C layout: vgpr v, lane -> D[m = v + 8*half][n = tile + mcol]
        // relu + segment_max == u32 atomicMax of positive values into 0-init output.
        // Single guard per value: zero/negative values (incl. padded rows) skip.
#pragma unroll
        for (int v = 0; v < 8; ++v) {
            const int m    = v + half * 8;
            const int segm = __shfl(seg, m, 32);
            unsigned int* basep = (unsigned int*)(out + (long)segm * COUT);
            const float lo = c_lo[v];
            const float hi = c_hi[v];
            if (lo > 0.f) atomicMax(basep + col_lo, __float_as_uint(lo));
            if (hi > 0.f) atomicMax(basep + col_hi, __float_as_uint(hi));
        }
    }
}

extern "C" void kernel_launch(void* const* d_in, const int* in_sizes, int n_in,
                              void* d_out, int out_size, void* d_ws, size_t ws_size,
                              hipStream_t stream) {
    const float* xyz       = (const float*)d_in[0];
    const int*   batch_cnt = (const int*)d_in[1];
    const float* pfeat     = (const float*)d_in[2];
    const float* W         = (const float*)d_in[3];
    float* out = (float*)d_out;

    const int N = in_sizes[0] / 3;
    const int num_groups = (N + 15) / 16;

    // 1) zero-init dense BEV output (atomicMax identity for relu'd values)
    const int n4 = out_size / 4;
    zero_f4_kernel<<<8192, 256, 0, stream>>>((float4*)out, n4);

    // 2) WMMA featurize + MLP + scatter-max
    pillar_wmma_kernel<<<2048, 256, 0, stream>>>(xyz, batch_cnt, pfeat, W, out,
                                                 N, num_groups);
}